// RNNGCN_5265629904970
// MI455X (gfx1250) — compile-verified
//
#include <hip/hip_runtime.h>

// Problem constants (match reference)
#define N_      4096
#define T_      10
#define F_      512
#define NHID_   64
#define NCLASS_ 16
#define NN_     ((size_t)N_ * N_)

typedef float v2f __attribute__((ext_vector_type(2)));
typedef float v8f __attribute__((ext_vector_type(8)));

// D = A(16x4,f32) * B(4x16,f32) + C(16x16,f32)
__device__ __forceinline__ v8f wmma4(v2f a, v2f b, v8f c) {
    return __builtin_amdgcn_wmma_f32_16x16x4_f32(
        /*neg_a=*/false, a, /*neg_b=*/false, b,
        /*c_mod=*/(short)0, c, /*reuse_a=*/false, /*reuse_b=*/false);
}

// CDNA5 async global->LDS copy (GV mode), tracked by ASYNCcnt.
__device__ __forceinline__ void async_b128(const float* l, const float* g) {
    unsigned lo = (unsigned)(size_t)l;   // flat shared addr [31:0] == LDS offset
    asm volatile("global_load_async_to_lds_b128 %0, %1, off"
                 :: "v"(lo), "v"(g)
                 : "memory");
}
__device__ __forceinline__ void wait_async0() {
    asm volatile("s_wait_asynccnt 0" ::: "memory");
}

// fold weights: A = w[0]*adj[0] + ... + w[9]*adj[9]
// w[0] = (1-l)^9 ; w[t] = l*(1-l)^(9-t) for t>=1
__device__ __forceinline__ void fold_weights(float lam, float* w) {
    float cc = 1.0f - lam;
    float p = 1.0f;
#pragma unroll
    for (int t = T_ - 1; t >= 1; --t) { w[t] = lam * p; p *= cc; }
    w[0] = p;
}

// ---------------------------------------------------------------------------
// Kernel 1: XW1t = (x @ W1)^T, x = feats[:, T-1, :]   -> XW1t is 64 x 4096
// block: 256 thr (8 waves), tile 32 rows x 64 cols, Kblk=64
// ---------------------------------------------------------------------------
__global__ __launch_bounds__(256) void k_xw1(const float* __restrict__ feats,
                                             const float* __restrict__ W1,
                                             float* __restrict__ XW1t) {
    __shared__ float sA[32 * 68];   // x tile, padded stride 68 (mod 64 == 4)
    __shared__ float sB[64 * 80];   // W1 tile (input row-major), stride 80
    const int tid = threadIdx.x;
    const int rowBase = blockIdx.x * 32;
    const int lane = tid & 31, wid = tid >> 5;
    const int rg = wid >> 2, cg = wid & 3;          // row group / col group
    const int m = lane & 15;
    const int kh = (lane >> 4) << 1;                // 0 or 2

    v8f c = {};
    for (int kb = 0; kb < F_; kb += 64) {
        __syncthreads();
#pragma unroll
        for (int ch = 0; ch < 4; ++ch) {            // W1 tile: async copy, 1024 f4 / 256 thr
            int f4 = tid + ch * 256;
            int kr = f4 >> 4, col = (f4 & 15) << 2;
            async_b128(sB + kr * 80 + col, W1 + (size_t)(kb + kr) * NHID_ + col);
        }
#pragma unroll
        for (int ch = 0; ch < 2; ++ch) {            // x tile: 512 float4 / 256 thr
            int f4 = tid + ch * 256;
            int row = f4 >> 4, col = (f4 & 15) << 2;
            float4 v = *(const float4*)(feats + (size_t)(rowBase + row) * (T_ * F_)
                                              + (size_t)(T_ - 1) * F_ + kb + col);
            *(float4*)(sA + row * 68 + col) = v;
        }
        wait_async0();
        __syncthreads();
#pragma unroll
        for (int kk = 0; kk < 64; kk += 4) {
            v2f a = *(const v2f*)(sA + (rg * 16 + m) * 68 + kk + kh);
            v2f b;
            b.x = sB[(kk + kh) * 80 + cg * 16 + m];
            b.y = sB[(kk + kh + 1) * 80 + cg * 16 + m];
            c = wmma4(a, b, c);
        }
    }
#pragma unroll
    for (int r = 0; r < 8; ++r) {
        int row = rowBase + rg * 16 + r + ((lane >> 4) << 3);
        int col = cg * 16 + m;
        XW1t[(size_t)col * N_ + row] = c[r];        // transposed store (64 x 4096)
    }
}

// ---------------------------------------------------------------------------
// Kernel 2: H = relu(A @ XW1 + b1), A folded on the fly from adj (671MB pass)
// B operand staged from transposed XW1t -> single ds_load_b64 fragments
// block: 256 thr, tile 32 rows x 64 cols, Kblk=64, K=4096
// ---------------------------------------------------------------------------
__global__ __launch_bounds__(256) void k_h(const float* __restrict__ adj,
                                           const float* __restrict__ lamp,
                                           const float* __restrict__ XW1t,
                                           const float* __restrict__ b1,
                                           float* __restrict__ H) {
    __shared__ float sA[32 * 68];   // folded A tile (rows x k), stride 68
    __shared__ float sB[64 * 68];   // XW1t tile (ncol x k), stride 68
    float w[T_];
    fold_weights(*lamp, w);
    const int tid = threadIdx.x;
    const int rowBase = blockIdx.x * 32;
    const int lane = tid & 31, wid = tid >> 5;
    const int rg = wid >> 2, cg = wid & 3;
    const int m = lane & 15;
    const int kh = (lane >> 4) << 1;

    v8f c = {};
    for (int kb = 0; kb < N_; kb += 64) {
        __syncthreads();
#pragma unroll
        for (int ch = 0; ch < 4; ++ch) {            // XW1t strip: async copy (64 rows x 64 k)
            int f4 = tid + ch * 256;
            int row = f4 >> 4, col = (f4 & 15) << 2;
            async_b128(sB + row * 68 + col, XW1t + (size_t)row * N_ + kb + col);
        }
#pragma unroll
        for (int ch = 0; ch < 2; ++ch) {            // fold 32x64 A tile (needs VALU)
            int f4 = tid + ch * 256;
            int row = f4 >> 4, col = (f4 & 15) << 2;
            size_t off = (size_t)(rowBase + row) * N_ + kb + col;
            float4 acc = {0.f, 0.f, 0.f, 0.f};
#pragma unroll
            for (int t = 0; t < T_; ++t) {
                const float* p = adj + (size_t)t * NN_ + off;
                float4 v = *(const float4*)p;
                acc.x += w[t] * v.x; acc.y += w[t] * v.y;
                acc.z += w[t] * v.z; acc.w += w[t] * v.w;
                if (kb + 64 < N_) __builtin_prefetch(p + 64, 0, 0);  // next K block
            }
            *(float4*)(sA + row * 68 + col) = acc;
        }
        wait_async0();
        __syncthreads();
#pragma unroll
        for (int kk = 0; kk < 64; kk += 4) {
            v2f a = *(const v2f*)(sA + (rg * 16 + m) * 68 + kk + kh);
            v2f b = *(const v2f*)(sB + (cg * 16 + m) * 68 + kk + kh);
            c = wmma4(a, b, c);
        }
    }
#pragma unroll
    for (int r = 0; r < 8; ++r) {
        int row = rowBase + rg * 16 + r + ((lane >> 4) << 3);
        int col = cg * 16 + m;
        float v = c[r] + b1[col];
        H[(size_t)row * NHID_ + col] = v > 0.f ? v : 0.f;
    }
}

// ---------------------------------------------------------------------------
// Kernel 3: HW2t = (H @ W2)^T   -> HW2t is 16 x 4096; K=64 single block
// block: 128 thr (4 waves), tile 64 rows x 16 cols
// ---------------------------------------------------------------------------
__global__ __launch_bounds__(128) void k_hw2(const float* __restrict__ H,
                                             const float* __restrict__ W2,
                                             float* __restrict__ HW2t) {
    __shared__ float sA[64 * 68];
    __shared__ float sB[64 * 20];
    const int tid = threadIdx.x;
    const int rowBase = blockIdx.x * 64;
    const int lane = tid & 31, wid = tid >> 5;      // wid == row group
    const int m = lane & 15;
    const int kh = (lane >> 4) << 1;

#pragma unroll
    for (int ch = 0; ch < 8; ++ch) {                // H tile: async copy, 1024 f4 / 128 thr
        int f4 = tid + ch * 128;
        int row = f4 >> 4, col = (f4 & 15) << 2;
        async_b128(sA + row * 68 + col, H + (size_t)(rowBase + row) * NHID_ + col);
    }
#pragma unroll
    for (int ch = 0; ch < 2; ++ch) {                // W2 (input row-major): async copy
        int f4 = tid + ch * 128;
        int kr = f4 >> 2, col = (f4 & 3) << 2;
        async_b128(sB + kr * 20 + col, W2 + (size_t)kr * NCLASS_ + col);
    }
    wait_async0();
    __syncthreads();
    v8f c = {};
    const float* pA = sA + (wid * 16 + m) * 68;
#pragma unroll
    for (int kk = 0; kk < 64; kk += 4) {
        v2f a = *(const v2f*)(pA + kk + kh);
        v2f b;
        b.x = sB[(kk + kh) * 20 + m];
        b.y = sB[(kk + kh + 1) * 20 + m];
        c = wmma4(a, b, c);
    }
#pragma unroll
    for (int r = 0; r < 8; ++r) {
        int row = rowBase + wid * 16 + r + ((lane >> 4) << 3);
        HW2t[(size_t)m * N_ + row] = c[r];          // transposed store (16 x 4096)
    }
}

// ---------------------------------------------------------------------------
// Kernel 4: out = softmax(A @ HW2 + b2), fold-on-the-fly (second 671MB pass)
// block: 256 thr (8 waves) = 2 row-groups x 4 K-partitions; tile 32 rows x 16 cols
// ---------------------------------------------------------------------------
__global__ __launch_bounds__(256) void k_out(const float* __restrict__ adj,
                                             const float* __restrict__ lamp,
                                             const float* __restrict__ HW2t,
                                             const float* __restrict__ b2,
                                             float* __restrict__ out) {
    __shared__ float sA[4 * 32 * 68];   // per-partition folded A tiles (also reused for C reduce)
    __shared__ float sB[4 * 16 * 68];   // per-partition HW2t strips (class x k)
    float w[T_];
    fold_weights(*lamp, w);
    const int tid = threadIdx.x;
    const int rowBase = blockIdx.x * 32;
    const int lane = tid & 31, wid = tid >> 5;
    const int rg = wid & 1, kp = wid >> 1;          // row group / K partition
    const int m = lane & 15;
    const int kh = (lane >> 4) << 1;

    v8f c = {};
    for (int it = 0; it < 16; ++it) {               // each partition covers 1024 K, 64/iter
        __syncthreads();
#pragma unroll
        for (int ch = 0; ch < 4; ++ch) {            // HW2t strips: async copy, 1024 f4 / 256 thr
            int f4 = tid + ch * 256;
            int part = f4 >> 8;
            int rem = f4 & 255;
            int row = rem >> 4, col = (rem & 15) << 2;   // row = class 0..15, col = k
            async_b128(sB + part * (16 * 68) + row * 68 + col,
                       HW2t + (size_t)row * N_ + part * 1024 + it * 64 + col);
        }
#pragma unroll
        for (int ch = 0; ch < 8; ++ch) {            // 2048 float4 / 256 thr (fold)
            int f4 = tid + ch * 256;
            int part = f4 >> 9;
            int rem = f4 & 511;
            int row = rem >> 4, col = (rem & 15) << 2;
            size_t off = (size_t)(rowBase + row) * N_ + part * 1024 + it * 64 + col;
            float4 acc = {0.f, 0.f, 0.f, 0.f};
#pragma unroll
            for (int t = 0; t < T_; ++t) {
                const float* p = adj + (size_t)t * NN_ + off;
                float4 v = *(const float4*)p;
                acc.x += w[t] * v.x; acc.y += w[t] * v.y;
                acc.z += w[t] * v.z; acc.w += w[t] * v.w;
                if (it < 15) __builtin_prefetch(p + 64, 0, 0);  // next K block
            }
            *(float4*)(sA + part * (32 * 68) + row * 68 + col) = acc;
        }
        wait_async0();
        __syncthreads();
        const float* pA = sA + kp * (32 * 68) + (rg * 16 + m) * 68;
        const float* pB = sB + kp * (16 * 68) + m * 68;
#pragma unroll
        for (int kk = 0; kk < 64; kk += 4) {
            v2f a = *(const v2f*)(pA + kk + kh);
            v2f b = *(const v2f*)(pB + kk + kh);
            c = wmma4(a, b, c);
        }
    }
    // reduce 4 K-partition partials per 16x16 tile, then bias + softmax
    __syncthreads();
    float* sC = sA;                                 // 8 waves * 16*16 = 2048 floats
#pragma unroll
    for (int r = 0; r < 8; ++r) {
        int row = r + ((lane >> 4) << 3);
        sC[wid * 256 + row * 16 + m] = c[r];
    }
    __syncthreads();
    if (tid < 32) {
        int rgl = tid >> 4, lr = tid & 15;          // output row = rgl*16 + lr = tid
        float logit[NCLASS_];
#pragma unroll
        for (int col = 0; col < NCLASS_; ++col) {
            float s = 0.f;
#pragma unroll
            for (int p = 0; p < 4; ++p)
                s += sC[(p * 2 + rgl) * 256 + lr * 16 + col];
            logit[col] = s + b2[col];
        }
        float mx = logit[0];
#pragma unroll
        for (int col = 1; col < NCLASS_; ++col) mx = fmaxf(mx, logit[col]);
        float sum = 0.f;
#pragma unroll
        for (int col = 0; col < NCLASS_; ++col) { logit[col] = __expf(logit[col] - mx); sum += logit[col]; }
        float inv = 1.0f / sum;
#pragma unroll
        for (int col = 0; col < NCLASS_; ++col)
            out[(size_t)(rowBase + tid) * NCLASS_ + col] = logit[col] * inv;
    }
}

// ---------------------------------------------------------------------------
extern "C" void kernel_launch(void* const* d_in, const int* in_sizes, int n_in,
                              void* d_out, int out_size, void* d_ws, size_t ws_size,
                              hipStream_t stream) {
    const float* feats = (const float*)d_in[0];
    const float* adj   = (const float*)d_in[1];
    const float* lam   = (const float*)d_in[2];
    const float* W1    = (const float*)d_in[3];
    const float* b1    = (const float*)d_in[4];
    const float* W2    = (const float*)d_in[5];
    const float* b2    = (const float*)d_in[6];
    float* outp = (float*)d_out;

    float* XW1t = (float*)d_ws;                 // 64 x 4096 f32 = 1 MB
    float* H    = XW1t + (size_t)NHID_ * N_;    // 4096 x 64 f32 = 1 MB
    float* HW2t = H + (size_t)N_ * NHID_;       // 16 x 4096 f32 = 256 KB

    k_xw1<<<N_ / 32, 256, 0, stream>>>(feats, W1, XW1t);
    k_h  <<<N_ / 32, 256, 0, stream>>>(adj, lam, XW1t, b1, H);
    k_hw2<<<N_ / 64, 128, 0, stream>>>(H, W2, HW2t);
    k_out<<<N_ / 32, 256, 0, stream>>>(adj, lam, HW2t, b2, outp);
}